// BRPConvEmbedding_84679575208613
// MI455X (gfx1250) — compile-verified
//
#include <hip/hip_runtime.h>
#include <hip/hip_bf16.h>

// ---- problem constants (match reference) ----
#define N_NODES  50000
#define N_EDGES  800000
#define DIM      128
#define N_LAYERS 3
#define N_GRAPHS 256

typedef __attribute__((ext_vector_type(2))) float v2f;
typedef __attribute__((ext_vector_type(8))) float v8f;

// ---------------- utility: zero a float region ----------------
__global__ void zero_f32(float* __restrict__ p, int n) {
    int t = blockIdx.x * blockDim.x + threadIdx.x;
    if (t < n) p[t] = 0.0f;
}

// ---------------- degree accumulation (float, native f32 atomics) ------
__global__ void degree_kernel(const int* __restrict__ src, const int* __restrict__ dst,
                              float* __restrict__ deg_out, float* __restrict__ deg_in,
                              int n_edges) {
    int e = blockIdx.x * blockDim.x + threadIdx.x;
    if (e >= n_edges) return;
    unsafeAtomicAdd(&deg_out[src[e]], 1.0f);
    unsafeAtomicAdd(&deg_in[dst[e]], 1.0f);
}

// ---------------- deg -> rsqrt(max(deg,1)) in place ----------------
__global__ void rsqrt_kernel(float* __restrict__ a, float* __restrict__ b, int n) {
    int t = blockIdx.x * blockDim.x + threadIdx.x;
    if (t >= n) return;
    a[t] = __frsqrt_rn(fmaxf(a[t], 1.0f));
    b[t] = __frsqrt_rn(fmaxf(b[t], 1.0f));
}

// ---------------- SpMM: agg[dst] += h[src] * norm_out[src] -------------
// One wave per edge; each lane owns a float4 chunk of the 128-wide row.
// Feature matrix (25.6MB) is L2-resident -> atomics resolve in L2.
__global__ void spmm_kernel(const float* __restrict__ h, const float* __restrict__ norm_out,
                            const int* __restrict__ src, const int* __restrict__ dst,
                            float* __restrict__ agg, int n_edges) {
    long long t = (long long)blockIdx.x * blockDim.x + threadIdx.x;
    int e = (int)(t >> 5);
    int c = (int)(t & 31);
    if (e >= n_edges) return;
    int s = src[e];
    int d = dst[e];
    float ns = norm_out[s];
    const float4 v = *reinterpret_cast<const float4*>(h + (size_t)s * DIM + c * 4);
    float* p = agg + (size_t)d * DIM + c * 4;
    unsafeAtomicAdd(p + 0, v.x * ns);
    unsafeAtomicAdd(p + 1, v.y * ns);
    unsafeAtomicAdd(p + 2, v.z * ns);
    unsafeAtomicAdd(p + 3, v.w * ns);
}

// ---------------- fused: out = relu((agg * norm_in) @ W + b) ----------
// Block = 256 threads = 8 waves. Block owns one 16-row M tile; wave w owns
// N-tile w (16 cols). A rows staged (pre-scaled by norm_in) into LDS with
// padded stride 132 (132 mod 64 = 4 -> 16 rows land in 16 distinct banks).
// K=128 done as 32 x V_WMMA_F32_16X16X4_F32. Safe in-place (reads complete
// into LDS before the barrier; writes touch only this block's rows).
__launch_bounds__(256, 2)
__global__ void gemm_relu_wmma(const float* __restrict__ A,
                               const float* __restrict__ norm_in,
                               const float* __restrict__ Wl,   // [128,128] (k, n)
                               const float* __restrict__ bias, // [128]
                               float* __restrict__ Out,
                               int n_rows) {
    __shared__ float sA[16 * 132];
    const int m0 = blockIdx.x * 16;
    const int tid = threadIdx.x;

    // stage 16 scaled rows (zero-pad OOB so WMMA runs with full EXEC)
    for (int i = tid; i < 16 * DIM; i += 256) {
        int r = i >> 7;
        int c = i & 127;
        int row = m0 + r;
        float v = 0.0f;
        if (row < n_rows) v = A[(size_t)row * DIM + c] * norm_in[row];
        sA[r * 132 + c] = v;
    }
    __syncthreads();

    const int wave = tid >> 5;   // N-tile index 0..7
    const int lane = tid & 31;
    const int hl = lane & 15;    // M row (A) / N col (B,C)
    const int hi = lane >> 4;    // K half-select (A,B) / M half-select (C)
    const int col = wave * 16 + hl;

    v8f acc = {};
#pragma unroll
    for (int kk = 0; kk < 32; ++kk) {
        const int kb = kk * 4 + hi * 2;
        v2f a, b;
        // A 16x4 f32: lanes 0-15 -> K {kb,kb+1}, lanes 16-31 -> K {4kk+2,4kk+3}
        a.x = sA[hl * 132 + kb + 0];
        a.y = sA[hl * 132 + kb + 1];
        // B 4x16 f32: row-striped across lanes, mirrored K split
        b.x = Wl[(size_t)(kb + 0) * DIM + col];
        b.y = Wl[(size_t)(kb + 1) * DIM + col];
        acc = __builtin_amdgcn_wmma_f32_16x16x4_f32(
            /*neg_a=*/false, a, /*neg_b=*/false, b,
            /*c_mod=*/(short)0, acc, /*reuse_a=*/false, /*reuse_b=*/false);
    }

    const float bb = bias[col];
#pragma unroll
    for (int j = 0; j < 8; ++j) {
        // C/D layout: VGPR j holds M = j (lanes 0-15) / M = j+8 (lanes 16-31)
        int row = m0 + j + hi * 8;
        if (row < n_rows) {
            float v = acc[j] + bb;
            Out[(size_t)row * DIM + col] = v > 0.0f ? v : 0.0f;
        }
    }
}

// ---------------- pooling: atomic per-graph sum + count ----------------
__global__ void pool_accum(const float* __restrict__ h, const int* __restrict__ gid,
                           float* __restrict__ pooled, float* __restrict__ cnt,
                           int n_nodes) {
    long long t = (long long)blockIdx.x * blockDim.x + threadIdx.x;
    int nidx = (int)(t >> 5);
    int c = (int)(t & 31);
    if (nidx >= n_nodes) return;
    int g = gid[nidx];
    const float4 v = *reinterpret_cast<const float4*>(h + (size_t)nidx * DIM + c * 4);
    float* p = pooled + (size_t)g * DIM + c * 4;
    unsafeAtomicAdd(p + 0, v.x);
    unsafeAtomicAdd(p + 1, v.y);
    unsafeAtomicAdd(p + 2, v.z);
    unsafeAtomicAdd(p + 3, v.w);
    if (c == 0) unsafeAtomicAdd(&cnt[g], 1.0f);
}

__global__ void pool_div(float* __restrict__ pooled, const float* __restrict__ cnt) {
    int t = blockIdx.x * blockDim.x + threadIdx.x;  // over 256*128
    int g = t >> 7;
    pooled[t] /= fmaxf(cnt[g], 1.0f);
}

// ======================================================================
extern "C" void kernel_launch(void* const* d_in, const int* in_sizes, int n_in,
                              void* d_out, int out_size, void* d_ws, size_t ws_size,
                              hipStream_t stream) {
    const float* feats = (const float*)d_in[0];   // [N_NODES, 128]
    const float* Wall  = (const float*)d_in[1];   // [3, 128, 128]
    const float* ball  = (const float*)d_in[2];   // [3, 128]
    const int*   src   = (const int*)d_in[3];     // [N_EDGES]
    const int*   dst   = (const int*)d_in[4];     // [N_EDGES]
    const int*   gid   = (const int*)d_in[5];     // [N_NODES]
    float* out = (float*)d_out;                   // [256, 128]

    float* ws = (float*)d_ws;
    float* norm_out = ws;                                       // N_NODES
    float* norm_in  = ws + N_NODES;                             // N_NODES
    float* bufA     = ws + 2 * (size_t)N_NODES;                 // N_NODES*DIM
    float* bufB     = bufA + (size_t)N_NODES * DIM;             // N_NODES*DIM
    float* cnt      = bufB + (size_t)N_NODES * DIM;             // N_GRAPHS

    const int BT = 256;
    // --- degree norms (computed once) ---
    zero_f32<<<(2 * N_NODES + BT - 1) / BT, BT, 0, stream>>>(norm_out, 2 * N_NODES);
    degree_kernel<<<(N_EDGES + BT - 1) / BT, BT, 0, stream>>>(src, dst, norm_out, norm_in, N_EDGES);
    rsqrt_kernel<<<(N_NODES + BT - 1) / BT, BT, 0, stream>>>(norm_out, norm_in, N_NODES);

    // --- 3 GraphConv layers (ping-pong; gemm is in-place on agg buffer) ---
    const float* hin = feats;
    float* bufs[2] = {bufA, bufB};
    int cur = 0;
    const long long spmm_threads = (long long)N_EDGES * 32;
    for (int l = 0; l < N_LAYERS; ++l) {
        float* agg = bufs[cur];
        zero_f32<<<(N_NODES * DIM + BT - 1) / BT, BT, 0, stream>>>(agg, N_NODES * DIM);
        spmm_kernel<<<(int)((spmm_threads + BT - 1) / BT), BT, 0, stream>>>(
            hin, norm_out, src, dst, agg, N_EDGES);
        gemm_relu_wmma<<<(N_NODES + 15) / 16, BT, 0, stream>>>(
            agg, norm_in, Wall + (size_t)l * DIM * DIM, ball + (size_t)l * DIM, agg, N_NODES);
        hin = agg;
        cur ^= 1;
    }

    // --- mean pooling readout ---
    zero_f32<<<(N_GRAPHS * DIM + BT - 1) / BT, BT, 0, stream>>>(out, N_GRAPHS * DIM);
    zero_f32<<<1, N_GRAPHS, 0, stream>>>(cnt, N_GRAPHS);
    const long long pool_threads = (long long)N_NODES * 32;
    pool_accum<<<(int)((pool_threads + BT - 1) / BT), BT, 0, stream>>>(hin, gid, out, cnt, N_NODES);
    pool_div<<<(N_GRAPHS * DIM) / BT, BT, 0, stream>>>(out, cnt);
}